// PagedAttention_69312182223772
// MI455X (gfx1250) — compile-verified
//
#include <hip/hip_runtime.h>
#include <hip/hip_bf16.h>
#include <hip/hip_fp16.h>

// ---------------------------------------------------------------------------
// Paged attention decode (GQA 4:1), f32, flash-style, V_WMMA_F32_16X16X4_F32.
//
// grid  = (NUM_KV_HEADS, NUM_SEQS), block = 128 threads (4 wave32)
// Each wave flash-decodes KV blocks {wave, wave+4, ...}. WMMA B-operands for
// both K and V are loaded straight from global memory (each lane's operand
// stream is a disjoint partition of the 2048-f32 K/V block, so no LDS staging
// is needed). Waves combine partial results via LDS log-sum-exp at the end.
// ---------------------------------------------------------------------------

typedef __attribute__((ext_vector_type(2))) float v2f;
typedef __attribute__((ext_vector_type(8))) float v8f;

#define PA_NUM_SEQS      64
#define PA_NUM_HEADS     32
#define PA_NUM_KV_HEADS  8
#define PA_HEAD          128
#define PA_BS            16
#define PA_MAXB          128
#define PA_SCALE         0.08838834764831845f
#define PA_NEG           -3.0e38f

__global__ __launch_bounds__(128) void PagedAttention_69312182223772_kernel(
    const float* __restrict__ qry,      // [64][32][128]
    const float* __restrict__ kcache,   // [8192][8][16][16][8]  (2048 f32 per (blk,kvh))
    const float* __restrict__ vcache,   // [8192][8][128][16]    (2048 f32 per (blk,kvh))
    const int*   __restrict__ btab,     // [64][128]
    const int*   __restrict__ ctxlen,   // [64]
    float*       __restrict__ out)      // [64][32*128]
{
    const int g    = blockIdx.x;          // kv head
    const int s    = blockIdx.y;          // sequence
    const int tid  = threadIdx.x;
    const int wave = tid >> 5;
    const int lane = tid & 31;
    const int ln   = lane & 15;           // column / row-low index inside 16-lane group
    const int hi   = lane >> 4;           // which 16-lane half

    // Only the P lane-transpose tile and the cross-wave partials live in LDS.
    __shared__ __align__(16) float pbuf[4][256];
    __shared__ float partO[4][4][128];
    __shared__ float partM[4][4];
    __shared__ float partL[4][4];

    const int ctx     = ctxlen[s];
    const int nblocks = (ctx + PA_BS - 1) >> 4;

    // ---- Load Q once into WMMA A-matrix chunk layout ----------------------
    // Chunk c covers dims 4c..4c+3. Lane<16: K=4c,4c+1 ; lane>=16: K=4c+2,4c+3.
    // Rows (M) 0..3 = the 4 query heads of this kv head; rows 4..15 = 0.
    v2f aq[32];
    {
        const float* qp = qry + ((size_t)s * PA_NUM_HEADS + (size_t)g * 4) * PA_HEAD;
        #pragma unroll
        for (int c = 0; c < 32; ++c) {
            float a0 = 0.0f, a1 = 0.0f;
            if (ln < 4) {
                const int d0 = 4 * c + 2 * hi;
                a0 = qp[ln * PA_HEAD + d0];
                a1 = qp[ln * PA_HEAD + d0 + 1];
            }
            aq[c][0] = a0;
            aq[c][1] = a1;
        }
    }

    // ---- Running flash state ---------------------------------------------
    v8f oacc[8];                       // output C-tiles: d-tile dt, rows r / r+8
    #pragma unroll
    for (int dt = 0; dt < 8; ++dt)
        #pragma unroll
        for (int i = 0; i < 8; ++i) oacc[dt][i] = 0.0f;

    float m_run[8], l_run[8];
    #pragma unroll
    for (int r = 0; r < 8; ++r) { m_run[r] = PA_NEG; l_run[r] = 0.0f; }

    float* pb = pbuf[wave];

    for (int blk = wave; blk < nblocks; blk += 4) {
        const int    phys  = btab[s * PA_MAXB + blk];
        const float* kp    = kcache + ((size_t)phys * PA_NUM_KV_HEADS + g) * 2048;
        const float* vp    = vcache + ((size_t)phys * PA_NUM_KV_HEADS + g) * 2048;

        // ---- S = Q * K^T : B-operands straight from global ----------------
        // K element (d,t) at offset (d>>3)*128 + t*8 + (d&7); chunk c needs
        // d0 = 4c + 2*hi at t = ln  ->  v2f at (c>>1)*128 + ln*8 + 4*(c&1) + 2*hi
        v8f sc0, sc1;
        #pragma unroll
        for (int i = 0; i < 8; ++i) { sc0[i] = 0.0f; sc1[i] = 0.0f; }

        #pragma unroll
        for (int half = 0; half < 2; ++half) {
            v2f kb[16];
            #pragma unroll
            for (int i = 0; i < 16; ++i) {
                const int c = half * 16 + i;
                kb[i] = __builtin_nontemporal_load(
                    (const v2f*)(kp + (c >> 1) * 128 + ln * 8 + 4 * (c & 1) + 2 * hi));
            }
            #pragma unroll
            for (int i = 0; i < 16; ++i) {
                const int c = half * 16 + i;
                if (i & 1)
                    sc1 = __builtin_amdgcn_wmma_f32_16x16x4_f32(
                            false, aq[c], false, kb[i], (short)0, sc1, false, false);
                else
                    sc0 = __builtin_amdgcn_wmma_f32_16x16x4_f32(
                            false, aq[c], false, kb[i], (short)0, sc0, false, false);
            }
        }
        const v8f sc = sc0 + sc1;

        // ---- Masked flash softmax tile update ----------------------------
        const int tbase = blk * PA_BS;
        float alpha[8];
        #pragma unroll
        for (int r = 0; r < 8; ++r) {
            float sv = sc[r] * PA_SCALE;
            if (tbase + ln >= ctx) sv = -1.0e30f;

            // row max across the 16-lane group
            float mt = sv;
            mt = fmaxf(mt, __shfl_xor(mt, 1));
            mt = fmaxf(mt, __shfl_xor(mt, 2));
            mt = fmaxf(mt, __shfl_xor(mt, 4));
            mt = fmaxf(mt, __shfl_xor(mt, 8));

            const float mn = fmaxf(m_run[r], mt);
            const float al = __expf(m_run[r] - mn);
            const float pv = __expf(sv - mn);

            float rs = pv;
            rs += __shfl_xor(rs, 1);
            rs += __shfl_xor(rs, 2);
            rs += __shfl_xor(rs, 4);
            rs += __shfl_xor(rs, 8);

            l_run[r] = l_run[r] * al + rs;
            m_run[r] = mn;
            alpha[r] = al;

            // P tile (row q = r + 8*hi, col = ln) for the PV A-matrix reads
            pb[(r + 8 * hi) * 16 + ln] = pv;
        }

        #pragma unroll
        for (int dt = 0; dt < 8; ++dt)
            #pragma unroll
            for (int r = 0; r < 8; ++r)
                oacc[dt][r] *= alpha[r];

        // ---- O += P * V : V B-operands straight from global ---------------
        // V element (d,t) at offset d*16 + t; chunk (tc,dt) needs
        // t0 = 4tc + 2*hi at d = dt*16 + ln -> v2f at (dt*16+ln)*16 + t0
        #pragma unroll
        for (int tc = 0; tc < 4; ++tc) {
            const int t0 = 4 * tc + 2 * hi;
            v2f a;
            a[0] = pb[ln * 16 + t0];
            a[1] = pb[ln * 16 + t0 + 1];

            v2f vb[8];
            #pragma unroll
            for (int dt = 0; dt < 8; ++dt)
                vb[dt] = __builtin_nontemporal_load(
                    (const v2f*)(vp + (dt * 16 + ln) * 16 + t0));

            #pragma unroll
            for (int dt = 0; dt < 8; ++dt)
                oacc[dt] = __builtin_amdgcn_wmma_f32_16x16x4_f32(
                              false, a, false, vb[dt], (short)0, oacc[dt], false, false);
        }
    }

    // ---- Export per-wave partials (only rows 0..3 are real heads) --------
    if (hi == 0) {
        #pragma unroll
        for (int dt = 0; dt < 8; ++dt)
            #pragma unroll
            for (int r = 0; r < 4; ++r)
                partO[wave][r][dt * 16 + ln] = oacc[dt][r];
    }
    if (lane == 0) {
        #pragma unroll
        for (int r = 0; r < 4; ++r) {
            partM[wave][r] = m_run[r];
            partL[wave][r] = l_run[r];
        }
    }
    __syncthreads();

    // ---- Cross-wave log-sum-exp combine, one thread per head-dim ---------
    const int d = tid;   // 0..127
    #pragma unroll
    for (int qq = 0; qq < 4; ++qq) {
        const float M = fmaxf(fmaxf(partM[0][qq], partM[1][qq]),
                              fmaxf(partM[2][qq], partM[3][qq]));
        float L = 0.0f, O = 0.0f;
        #pragma unroll
        for (int w = 0; w < 4; ++w) {
            const float e = __expf(partM[w][qq] - M);
            L += e * partL[w][qq];
            O += e * partO[w][qq][d];
        }
        out[((size_t)s * PA_NUM_HEADS + (size_t)g * 4 + qq) * PA_HEAD + d] = O / L;
    }
}

extern "C" void kernel_launch(void* const* d_in, const int* in_sizes, int n_in,
                              void* d_out, int out_size, void* d_ws, size_t ws_size,
                              hipStream_t stream) {
    (void)in_sizes; (void)n_in; (void)out_size; (void)d_ws; (void)ws_size;
    const float* qry    = (const float*)d_in[0];
    const float* kcache = (const float*)d_in[1];
    const float* vcache = (const float*)d_in[2];
    const int*   btab   = (const int*)d_in[3];
    const int*   clen   = (const int*)d_in[4];
    float*       out    = (float*)d_out;

    dim3 grid(PA_NUM_KV_HEADS, PA_NUM_SEQS);
    dim3 block(128);
    PagedAttention_69312182223772_kernel<<<grid, block, 0, stream>>>(
        qry, kcache, vcache, btab, clen, out);
}